// HorizonReward_82145544503801
// MI455X (gfx1250) — compile-verified
//
#include <hip/hip_runtime.h>
#include <hip/hip_bf16.h>

typedef __attribute__((ext_vector_type(2))) float v2f;
typedef __attribute__((ext_vector_type(8))) float v8f;

namespace {
constexpr int   kNRbf = 50;
constexpr int   kH    = 60;
constexpr float kDT   = 0.05f;
constexpr float kJit  = 1e-6f;
constexpr float kPN   = 0.01f;
constexpr float kPML  = 0.05f, kG = 9.8f, kLEN = 0.5f, kMP = 0.1f, kMT = 1.1f;
}

// Single-wave latency-optimized rollout; policy GEMV runs on v_wmma_f32_16x16x4_f32.
__global__ __launch_bounds__(32)
void HorizonReward_kernel(const float* __restrict__ prm, float* __restrict__ out)
{
  const int  lane = (int)threadIdx.x;
  const bool lo   = lane < 16;

  __shared__ float Crows[64][16];   // quadratic-form coefficient rows, padded to 64x16

  // ---- One-time precompute: C[j] with E_j(x) = C[j] . [1, x, xx_upper, 0] = (x-mu_j)^T L_j L_j^T (x-mu_j)
  for (int rep = 0; rep < 2; ++rep) {
    const int j = lane + rep * 32;            // rows 0..63
    float row[16];
#pragma unroll
    for (int i = 0; i < 16; ++i) row[i] = 0.0f;
    if (j < kNRbf) {
      const float* sp = prm + 250 + j * 10;   // chol params for RBF j
      float L[4][4];
#pragma unroll
      for (int a = 0; a < 4; ++a)
#pragma unroll
        for (int b = 0; b < 4; ++b) L[a][b] = 0.0f;
      L[0][0] = sp[0]; L[1][1] = sp[1]; L[2][2] = sp[2]; L[3][3] = sp[3];
      L[1][0] = sp[4]; L[2][0] = sp[5]; L[2][1] = sp[6];
      L[3][0] = sp[7]; L[3][1] = sp[8]; L[3][2] = sp[9];
      float A[4][4];
#pragma unroll
      for (int a = 0; a < 4; ++a)
#pragma unroll
        for (int b = 0; b < 4; ++b) {
          float s = 0.0f;
#pragma unroll
          for (int k = 0; k < 4; ++k) s += L[a][k] * L[b][k];   // A = L L^T
          A[a][b] = s;
        }
      float mu[4];
#pragma unroll
      for (int i = 0; i < 4; ++i) mu[i] = prm[50 + i * 50 + j];
      float Amu[4];
#pragma unroll
      for (int i = 0; i < 4; ++i) {
        float s = 0.0f;
#pragma unroll
        for (int q = 0; q < 4; ++q) s += A[i][q] * mu[q];
        Amu[i] = s;
      }
      float c0 = 0.0f;
#pragma unroll
      for (int i = 0; i < 4; ++i) c0 += mu[i] * Amu[i];
      row[0] = c0;
#pragma unroll
      for (int i = 0; i < 4; ++i) row[1 + i] = -2.0f * Amu[i];
      int idx = 5;
#pragma unroll
      for (int a = 0; a < 4; ++a)
#pragma unroll
        for (int b = a; b < 4; ++b) row[idx++] = A[a][b] * ((a == b) ? 1.0f : 2.0f);
      // row[15] stays 0
    }
#pragma unroll
    for (int i = 0; i < 16; ++i) Crows[j][i] = row[i];
  }
  __syncthreads();

  // ---- Load WMMA A fragments (ISA 32-bit A 16x4 layout: lanes 0-15 K={0,1}, lanes 16-31 K={2,3})
  v2f   afrag[4][4];
  float wreg[4][8];
#pragma unroll
  for (int t = 0; t < 4; ++t) {
    const int row = t * 16 + (lane & 15);
#pragma unroll
    for (int k = 0; k < 4; ++k) {
      const int kk = k * 4 + (lo ? 0 : 2);
      v2f a; a.x = Crows[row][kk]; a.y = Crows[row][kk + 1];
      afrag[t][k] = a;
    }
    // D readout rows: lanes 0-15 -> M=r, lanes 16-31 -> M=8+r
#pragma unroll
    for (int r = 0; r < 8; ++r) {
      const int idx = t * 16 + (lo ? 0 : 8) + r;
      wreg[t][r] = (idx < kNRbf) ? prm[idx] : 0.0f;
    }
  }

  // ---- UT constants
  float Wt[9];
  Wt[0] = 0.2f;
#pragma unroll
  for (int i = 1; i < 9; ++i) Wt[i] = 0.1f;
  const float sqrt5   = sqrtf(5.0f);
  const float covDiag = kPN * kDT + kJit;       // exact analytic offset covariance
  const float invMT   = 1.0f / kMT;             // constant reciprocals (fold divides)
  const float pmlOverMT = kPML / kMT;
  const float mpOverMT  = kMP / kMT;

  // ---- Initial sigma points: x0=[0,0,0.1,0], chol(1e-6 I)
  float S[4][9];
  const float x0[4] = {0.0f, 0.0f, 0.1f, 0.0f};
  const float r0 = sqrt5 * sqrtf(kJit);
#pragma unroll
  for (int p = 0; p < 4; ++p)
#pragma unroll
    for (int i = 0; i < 9; ++i) S[p][i] = x0[p];
#pragma unroll
  for (int i = 0; i < 4; ++i) { S[i][1 + i] += r0; S[i][5 + i] -= r0; }

  float total = 0.0f;
#pragma unroll 1
  for (int h = 0; h < kH; ++h) {
    // mean = S @ Wt
    float m4[4];
#pragma unroll
    for (int p = 0; p < 4; ++p) {
      float s = 0.0f;
#pragma unroll
      for (int i = 0; i < 9; ++i) s += S[p][i] * Wt[i];
      m4[p] = s;
    }
    // feature vector f(x) = [1, x, upper(x x^T), 0]
    float f[16];
    f[0] = 1.0f;
#pragma unroll
    for (int i = 0; i < 4; ++i) f[1 + i] = m4[i];
    {
      int idx = 5;
#pragma unroll
      for (int a = 0; a < 4; ++a)
#pragma unroll
        for (int b = a; b < 4; ++b) f[idx++] = m4[a] * m4[b];
      f[15] = 0.0f;
    }
    // B fragments: broadcast f across all 16 N columns (N-mapping independent)
    v2f bfrag[4];
#pragma unroll
    for (int k = 0; k < 4; ++k) {
      const int kk = k * 4 + (lo ? 0 : 2);
      v2f b; b.x = f[kk]; b.y = f[kk + 1];
      bfrag[k] = b;
    }
    // E = C @ f :  4 M-tiles x 4 K-chunks of v_wmma_f32_16x16x4_f32
    float partial = 0.0f;
#pragma unroll
    for (int t = 0; t < 4; ++t) {
      v8f acc = {0.0f, 0.0f, 0.0f, 0.0f, 0.0f, 0.0f, 0.0f, 0.0f};
#pragma unroll
      for (int k = 0; k < 4; ++k)
        acc = __builtin_amdgcn_wmma_f32_16x16x4_f32(
            false, afrag[t][k], false, bfrag[k], (short)0, acc, false, false);
#pragma unroll
      for (int r = 0; r < 8; ++r) partial += wreg[t][r] * __expf(-acc[r]);
    }
    const float dotwphi = __shfl(partial, 0, 32) + __shfl(partial, 16, 32);
    const float u = fminf(10.0f, fmaxf(-10.0f, dotwphi));

    // cart-pole dynamics on the 9 sigma points (uniform across lanes)
    float nm[4][9];
#pragma unroll
    for (int i = 0; i < 9; ++i) {
      const float pos = S[0][i], vel = S[1][i], th = S[2][i], thd = S[3][i];
      const float sn = __sinf(th), cs = __cosf(th);
      const float temp  = (u + kPML * thd * thd * sn) * invMT;
      const float den   = kLEN * (4.0f / 3.0f - mpOverMT * cs * cs);
      const float thacc = (kG * sn - cs * temp) * __builtin_amdgcn_rcpf(den);
      const float xacc  = temp - pmlOverMT * thacc * cs;
      nm[0][i] = pos + kDT * vel;
      nm[1][i] = vel + kDT * xacc;
      nm[2][i] = th  + kDT * thd;
      nm[3][i] = thd + kDT * thacc;
    }

    // Analytic 81-point UT collapse (exact: offsets are symmetric, cross terms cancel):
    //   m    = nm @ Wt
    //   covc = sum_sig Wt_sig (nm_sig - m)(nm_sig - m)^T + (PN*DT + JITTER) I
    float mN[4];
#pragma unroll
    for (int p = 0; p < 4; ++p) {
      float s = 0.0f;
#pragma unroll
      for (int i = 0; i < 9; ++i) s += nm[p][i] * Wt[i];
      mN[p] = s;
    }
    float cc[4][4];
#pragma unroll
    for (int a = 0; a < 4; ++a)
#pragma unroll
      for (int b = 0; b < 4; ++b) cc[a][b] = 0.0f;
#pragma unroll
    for (int i = 0; i < 9; ++i) {
      float d4[4];
#pragma unroll
      for (int p = 0; p < 4; ++p) d4[p] = nm[p][i] - mN[p];
#pragma unroll
      for (int a = 0; a < 4; ++a)
#pragma unroll
        for (int b = a; b < 4; ++b) cc[a][b] += Wt[i] * d4[a] * d4[b];
    }
#pragma unroll
    for (int a = 0; a < 4; ++a) {
      cc[a][a] += covDiag + kJit;   // offset covariance diag + gen_sigma jitter
#pragma unroll
      for (int b = a + 1; b < 4; ++b) cc[b][a] = cc[a][b];
    }
    // 4x4 Cholesky
    float Lc[4][4];
#pragma unroll
    for (int a = 0; a < 4; ++a)
#pragma unroll
      for (int b = 0; b < 4; ++b) Lc[a][b] = 0.0f;
    for (int i = 0; i < 4; ++i)
      for (int j = 0; j <= i; ++j) {
        float s = cc[i][j];
        for (int k = 0; k < j; ++k) s -= Lc[i][k] * Lc[j][k];
        Lc[i][j] = (i == j) ? sqrtf(s) : s / Lc[j][j];
      }
    // regenerate sigma points
#pragma unroll
    for (int p = 0; p < 4; ++p) S[p][0] = mN[p];
#pragma unroll
    for (int c = 0; c < 4; ++c)
#pragma unroll
      for (int p = 0; p < 4; ++p) {
        const float rt = sqrt5 * Lc[p][c];
        S[p][1 + c] = mN[p] + rt;
        S[p][5 + c] = mN[p] - rt;
      }
    // reward on new states
    float rsum = 0.0f;
#pragma unroll
    for (int i = 0; i < 9; ++i)
      rsum += Wt[i] * (S[0][i] * S[0][i] + S[1][i] * S[1][i] +
                       10.0f * S[2][i] * S[2][i] + S[3][i] * S[3][i]);
    total -= rsum;
  }

  if (lane == 0) out[0] = total;
}

extern "C" void kernel_launch(void* const* d_in, const int* in_sizes, int n_in,
                              void* d_out, int out_size, void* d_ws, size_t ws_size,
                              hipStream_t stream) {
  (void)in_sizes; (void)n_in; (void)d_ws; (void)ws_size; (void)out_size;
  const float* prm = (const float*)d_in[0];   // 750 params, float32
  float* out = (float*)d_out;                 // scalar float32
  HorizonReward_kernel<<<1, 32, 0, stream>>>(prm, out);
}